// EdgeFeatureConvBlock_31473520345738
// MI455X (gfx1250) — compile-verified
//
#include <hip/hip_runtime.h>
#include <math.h>

#define Bv   32
#define Nv   1024
#define Pv   8192
#define Kv   16
#define Mv   (Nv * Kv)          // 16384 columns per batch for the conv layers
#define C0v  68                 // 2*C_IN + E_IN
#define EPSv 1e-5f

typedef float v2f __attribute__((ext_vector_type(2)));
typedef float v8f __attribute__((ext_vector_type(8)));

// ---------------------------------------------------------------------------
// WMMA tile engine: one wave computes a 64(out-ch) x 16(col) f32 tile of
// Y = W(64xCIN) @ X(CINx128-tile) using V_WMMA_F32_16X16X4_F32, K stepped by 4.
// A-frag layout (16x4 f32): lanes 0-15 -> M=0..15,K={k0,k0+1};
//                           lanes 16-31 -> M=0..15,K={k0+2,k0+3}.
// B-frag layout (4x16 f32): lanes 0-15 -> N=0..15,K={k0,k0+1};
//                           lanes 16-31 -> N=0..15,K={k0+2,k0+3}.
// C/D layout: VGPR v, lanes 0-15 -> M=v, lanes 16-31 -> M=v+8; N=lane&15.
// ---------------------------------------------------------------------------
template <int CIN, int WSTR>
__device__ __forceinline__ void wmma_64xC_tile(const float* Ws,   // LDS 64 x WSTR
                                               const float* Xs,   // LDS CIN x 128
                                               int colBase, int lane,
                                               v8f acc[4]) {
  const int cl = lane & 15;
  const int kh = (lane >> 4) << 1;  // 0 or 2
#pragma unroll
  for (int k0 = 0; k0 < CIN; k0 += 4) {
    v2f bf;
    bf.x = Xs[(k0 + kh + 0) * 128 + colBase + cl];
    bf.y = Xs[(k0 + kh + 1) * 128 + colBase + cl];
#pragma unroll
    for (int t = 0; t < 4; t++) {
      const int m = t * 16 + cl;
      v2f af;
      af.x = Ws[m * WSTR + k0 + kh + 0];
      af.y = Ws[m * WSTR + k0 + kh + 1];
      acc[t] = __builtin_amdgcn_wmma_f32_16x16x4_f32(false, af, false, bf,
                                                     (short)0, acc[t],
                                                     false, false);
    }
  }
}

// ---------------------------------------------------------------------------
// 0) zero the stats accumulators (4 sets x 128 floats)
// ---------------------------------------------------------------------------
__global__ void efc_zero_stats(float* stats) {
  int t = threadIdx.x;
  if (t < 512) stats[t] = 0.0f;
}

// ---------------------------------------------------------------------------
// 1) KNN over 2-D points: one thread per (b,n), top-17 smallest distances,
//    drop self (largest pd = -dist^2, stable tie order by index).
// ---------------------------------------------------------------------------
__global__ void __launch_bounds__(256) efc_knn(const float* __restrict__ pts,
                                               int* __restrict__ knn) {
  int gid = blockIdx.x * 256 + threadIdx.x;      // B*N threads
  if (gid >= Bv * Nv) return;
  int b = gid >> 10, n = gid & (Nv - 1);
  const float* px = pts + (size_t)(b * 2 + 0) * Nv;
  const float* py = pts + (size_t)(b * 2 + 1) * Nv;
  float xi = px[n], yi = py[n];
  float bd[17];
  int   bi[17];
#pragma unroll
  for (int t = 0; t < 17; t++) { bd[t] = -3.0e38f; bi[t] = -1; }
  for (int m = 0; m < Nv; m++) {
    float dx = px[m] - xi, dy = py[m] - yi;
    float pd = -(dx * dx + dy * dy);
    if (pd > bd[16]) {
      int pos = 16;
      while (pos > 0 && pd > bd[pos - 1]) {
        bd[pos] = bd[pos - 1]; bi[pos] = bi[pos - 1]; --pos;
      }
      bd[pos] = pd; bi[pos] = m;
    }
  }
#pragma unroll
  for (int k = 0; k < Kv; k++) knn[(size_t)gid * Kv + k] = bi[k + 1];
}

// ---------------------------------------------------------------------------
// 2) init edge tables: idxT <- N-1, efT <- 0
// ---------------------------------------------------------------------------
__global__ void __launch_bounds__(256) efc_edge_init(int* __restrict__ idxT,
                                                     float* __restrict__ efT) {
  int gid = blockIdx.x * 256 + threadIdx.x;      // covers B*4*N*K
  if (gid < Bv * 4 * Mv) efT[gid] = 0.0f;
  if (gid < Bv * Mv)     idxT[gid] = Nv - 1;
}

// ---------------------------------------------------------------------------
// 3) edge scatter: src sorted ascending -> j = i - lower_bound(src, src[i]).
//    (cutoff branch of the reference is dead for sorted non-negative src)
// ---------------------------------------------------------------------------
__global__ void __launch_bounds__(256) efc_edge_scatter(const int* __restrict__ el,
                                                        const float* __restrict__ ef,
                                                        int* __restrict__ idxT,
                                                        float* __restrict__ efT) {
  int gid = blockIdx.x * 256 + threadIdx.x;      // B*P threads
  if (gid >= Bv * Pv) return;
  int b = gid >> 13, i = gid & (Pv - 1);
  const int* src = el + (size_t)(b * 2 + 0) * Pv;
  int pf = src[i];
  int lo = 0, hi = Pv;
  while (lo < hi) { int mid = (lo + hi) >> 1; if (src[mid] < pf) lo = mid + 1; else hi = mid; }
  int j = i - lo;
  if (j < Kv) {
    int tgt = el[(size_t)(b * 2 + 1) * Pv + i];
    idxT[((size_t)b * Nv + pf) * Kv + j] = tgt;
#pragma unroll
    for (int c = 0; c < 4; c++)
      efT[(size_t)(b * 4 + c) * Mv + pf * Kv + j] = ef[(size_t)(b * 4 + c) * Pv + i];
  }
}

// ---------------------------------------------------------------------------
// 4) finalize neighbor index: fall back to knn where slot 0 untouched
// ---------------------------------------------------------------------------
__global__ void __launch_bounds__(256) efc_idx_final(const int* __restrict__ idxT,
                                                     const int* __restrict__ knn,
                                                     int* __restrict__ topIdx) {
  int gid = blockIdx.x * 256 + threadIdx.x;      // B*N*K threads
  if (gid >= Bv * Nv * Kv) return;
  int bn = gid >> 4;
  int i0 = idxT[(size_t)bn * Kv];
  topIdx[gid] = (i0 == Nv - 1) ? knn[gid] : idxT[gid];
}

// ---------------------------------------------------------------------------
// 5) layer-0 GEMM: X = [xi ; neigh-xi ; ef] (68 x cols) built in LDS by
//    gather, Y0 = W0 @ X, write y + per-channel sum/sumsq stats.
// ---------------------------------------------------------------------------
__global__ void __launch_bounds__(256) efc_conv0(const float* __restrict__ F,
                                                 const float* __restrict__ W0,
                                                 const int* __restrict__ topIdx,
                                                 const float* __restrict__ efT,
                                                 float* __restrict__ y0,
                                                 float* __restrict__ gSum,
                                                 float* __restrict__ gSq) {
  __shared__ float Xs[C0v * 128];
  __shared__ float Ws[64 * 69];
  __shared__ float sSum[64], sSq[64];
  int tid = threadIdx.x;
  int blk = blockIdx.x;                 // B * 128 blocks
  int b = blk >> 7;
  int colBase = (blk & 127) * 128;      // in [0, 16384)
  if (tid < 64) { sSum[tid] = 0.0f; sSq[tid] = 0.0f; }
  for (int e = tid; e < 64 * C0v; e += 256)
    Ws[(e / C0v) * 69 + (e % C0v)] = W0[e];
  {
    int col = tid >> 1, half = tid & 1;
    int j = colBase + col;
    int n = j >> 4;
    int idx = topIdx[(size_t)b * Mv + j];
    const float* Fb = F + (size_t)b * 32 * Nv;
#pragma unroll
    for (int cc = 0; cc < 16; cc++) {
      int c = half * 16 + cc;
      float xi = Fb[c * Nv + n];
      float xn = Fb[c * Nv + idx];
      Xs[c * 128 + col] = xi;
      Xs[(32 + c) * 128 + col] = xn - xi;
    }
    if (half == 0) {
#pragma unroll
      for (int e2 = 0; e2 < 4; e2++)
        Xs[(64 + e2) * 128 + col] = efT[(size_t)(b * 4 + e2) * Mv + j];
    }
  }
  __syncthreads();
  int lane = tid & 31, wave = tid >> 5;
  v8f acc[4] = {};
  wmma_64xC_tile<C0v, 69>(Ws, Xs, wave * 16, lane, acc);
  float* yb = y0 + (size_t)b * 64 * Mv;
  int cg = colBase + wave * 16 + (lane & 15);
  int mo = (lane >> 4) << 3;
#pragma unroll
  for (int t = 0; t < 4; t++)
#pragma unroll
    for (int v = 0; v < 8; v++) {
      int m = t * 16 + v + mo;
      float val = acc[t][v];
      yb[(size_t)m * Mv + cg] = val;
      atomicAdd(&sSum[m], val);
      atomicAdd(&sSq[m], val * val);
    }
  __syncthreads();
  if (tid < 64) { atomicAdd(&gSum[tid], sSum[tid]); atomicAdd(&gSq[tid], sSq[tid]); }
}

// ---------------------------------------------------------------------------
// 6) layers 1/2 GEMM: X = relu(bn(yIn)) applied on LDS stage, Y = W @ X
// ---------------------------------------------------------------------------
__global__ void __launch_bounds__(256) efc_conv12(const float* __restrict__ yIn,
                                                  const float* __restrict__ W,
                                                  const float* __restrict__ scl,
                                                  const float* __restrict__ bias,
                                                  float* __restrict__ yOut,
                                                  float* __restrict__ gSum,
                                                  float* __restrict__ gSq) {
  __shared__ float Xs[64 * 128];
  __shared__ float Ws[64 * 65];
  __shared__ float sSum[64], sSq[64];
  int tid = threadIdx.x;
  int blk = blockIdx.x;
  int b = blk >> 7;
  int colBase = (blk & 127) * 128;
  if (tid < 64) { sSum[tid] = 0.0f; sSq[tid] = 0.0f; }
  for (int e = tid; e < 64 * 64; e += 256)
    Ws[(e >> 6) * 65 + (e & 63)] = W[e];
  const float* yb = yIn + (size_t)b * 64 * Mv;
  for (int e = tid; e < 64 * 128; e += 256) {
    int c = e >> 7, col = e & 127;
    float v = yb[(size_t)c * Mv + colBase + col];
    v = v * scl[c] + bias[c];
    Xs[c * 128 + col] = v > 0.0f ? v : 0.0f;
  }
  __syncthreads();
  int lane = tid & 31, wave = tid >> 5;
  v8f acc[4] = {};
  wmma_64xC_tile<64, 65>(Ws, Xs, wave * 16, lane, acc);
  float* yo = yOut + (size_t)b * 64 * Mv;
  int cg = colBase + wave * 16 + (lane & 15);
  int mo = (lane >> 4) << 3;
#pragma unroll
  for (int t = 0; t < 4; t++)
#pragma unroll
    for (int v = 0; v < 8; v++) {
      int m = t * 16 + v + mo;
      float val = acc[t][v];
      yo[(size_t)m * Mv + cg] = val;
      atomicAdd(&sSum[m], val);
      atomicAdd(&sSq[m], val * val);
    }
  __syncthreads();
  if (tid < 64) { atomicAdd(&gSum[tid], sSum[tid]); atomicAdd(&gSq[tid], sSq[tid]); }
}

// ---------------------------------------------------------------------------
// 7) shortcut GEMM: sc = sc_w(64x32) @ features(32 x B*N), + stats
// ---------------------------------------------------------------------------
__global__ void __launch_bounds__(256) efc_sc(const float* __restrict__ F,
                                              const float* __restrict__ scW,
                                              float* __restrict__ sc,
                                              float* __restrict__ gSum,
                                              float* __restrict__ gSq) {
  __shared__ float Xs[32 * 128];
  __shared__ float Ws[64 * 33];
  __shared__ float sSum[64], sSq[64];
  int tid = threadIdx.x;
  int colBase = blockIdx.x * 128;       // 256 blocks over B*N columns
  int b = colBase >> 10;
  int nBase = colBase & (Nv - 1);
  if (tid < 64) { sSum[tid] = 0.0f; sSq[tid] = 0.0f; }
  for (int e = tid; e < 64 * 32; e += 256)
    Ws[(e >> 5) * 33 + (e & 31)] = scW[e];
  for (int e = tid; e < 32 * 128; e += 256) {
    int c = e >> 7, col = e & 127;
    Xs[c * 128 + col] = F[(size_t)(b * 32 + c) * Nv + nBase + col];
  }
  __syncthreads();
  int lane = tid & 31, wave = tid >> 5;
  v8f acc[4] = {};
  wmma_64xC_tile<32, 33>(Ws, Xs, wave * 16, lane, acc);
  int cg = nBase + wave * 16 + (lane & 15);
  int mo = (lane >> 4) << 3;
#pragma unroll
  for (int t = 0; t < 4; t++)
#pragma unroll
    for (int v = 0; v < 8; v++) {
      int m = t * 16 + v + mo;
      float val = acc[t][v];
      sc[(size_t)(b * 64 + m) * Nv + cg] = val;
      atomicAdd(&sSum[m], val);
      atomicAdd(&sSq[m], val * val);
    }
  __syncthreads();
  if (tid < 64) { atomicAdd(&gSum[tid], sSum[tid]); atomicAdd(&gSq[tid], sSq[tid]); }
}

// ---------------------------------------------------------------------------
// 8) BN finalize: scale = g*rsqrt(var+eps), bias = b - mean*scale
// ---------------------------------------------------------------------------
__global__ void efc_bn_finalize(const float* __restrict__ sum,
                                const float* __restrict__ sq,
                                const float* __restrict__ g,
                                const float* __restrict__ bvec,
                                float invCnt,
                                float* __restrict__ scale,
                                float* __restrict__ bias) {
  int o = threadIdx.x;
  if (o < 64) {
    float m = sum[o] * invCnt;
    float var = sq[o] * invCnt - m * m;
    float a = g[o] * rsqrtf(var + EPSv);
    scale[o] = a;
    bias[o] = bvec[o] - m * a;
  }
}

// ---------------------------------------------------------------------------
// 9) final fuse: out = relu( bn(sc) + mean_k relu(bn(y2)) )
// ---------------------------------------------------------------------------
__global__ void __launch_bounds__(256) efc_final(const float* __restrict__ y2,
                                                 const float* __restrict__ s2,
                                                 const float* __restrict__ b2,
                                                 const float* __restrict__ sc,
                                                 const float* __restrict__ sSc,
                                                 const float* __restrict__ bSc,
                                                 float* __restrict__ out) {
  int gid = blockIdx.x * 256 + threadIdx.x;   // B*64*N threads
  if (gid >= Bv * 64 * Nv) return;
  int n = gid & (Nv - 1);
  int bo = gid >> 10;                          // b*64 + o
  int o = bo & 63;
  float a2 = s2[o], c2 = b2[o];
  const float* yr = y2 + (size_t)bo * Mv + n * Kv;
  float accv = 0.0f;
#pragma unroll
  for (int k = 0; k < Kv; k++) {
    float v = yr[k] * a2 + c2;
    accv += v > 0.0f ? v : 0.0f;
  }
  float fts = accv * (1.0f / 16.0f);
  float scv = sc[(size_t)bo * Nv + n] * sSc[o] + bSc[o];
  float r = scv + fts;
  out[gid] = r > 0.0f ? r : 0.0f;
}

// ---------------------------------------------------------------------------
extern "C" void kernel_launch(void* const* d_in, const int* in_sizes, int n_in,
                              void* d_out, int out_size, void* d_ws, size_t ws_size,
                              hipStream_t stream) {
  (void)in_sizes; (void)n_in; (void)out_size; (void)ws_size;
  const float* points   = (const float*)d_in[0];
  const float* features = (const float*)d_in[1];
  const int*   edges    = (const int*)d_in[2];
  const float* edgef    = (const float*)d_in[3];
  const float* W0   = (const float*)d_in[5];
  const float* W1   = (const float*)d_in[6];
  const float* W2   = (const float*)d_in[7];
  const float* g0   = (const float*)d_in[8];
  const float* b0   = (const float*)d_in[9];
  const float* g1   = (const float*)d_in[10];
  const float* b1   = (const float*)d_in[11];
  const float* g2   = (const float*)d_in[12];
  const float* b2   = (const float*)d_in[13];
  const float* scW  = (const float*)d_in[14];
  const float* scG  = (const float*)d_in[15];
  const float* scB  = (const float*)d_in[16];
  float* out = (float*)d_out;

  char* ws = (char*)d_ws;
  size_t off = 0;
  float* yA   = (float*)(ws + off); off += (size_t)Bv * 64 * Mv * 4;   // 134 MB (y0, then y2)
  float* yB   = (float*)(ws + off); off += (size_t)Bv * 64 * Mv * 4;   // 134 MB (y1)
  float* efT  = (float*)(ws + off); off += (size_t)Bv * 4 * Mv * 4;    // 8 MB
  float* scBuf= (float*)(ws + off); off += (size_t)Bv * 64 * Nv * 4;   // 8 MB
  int*   knn  = (int*)(ws + off);   off += (size_t)Bv * Nv * Kv * 4;   // 2 MB
  int*   idxT = (int*)(ws + off);   off += (size_t)Bv * Nv * Kv * 4;   // 2 MB
  int*   topI = (int*)(ws + off);   off += (size_t)Bv * Nv * Kv * 4;   // 2 MB
  float* stats= (float*)(ws + off); off += 4 * 128 * 4;                // sum/sq x4 sets
  float* sb   = (float*)(ws + off); off += 4 * 128 * 4;                // scale/bias x4 sets

  float* sum0 = stats + 0 * 128;   float* sq0 = sum0 + 64;
  float* sum1 = stats + 1 * 128;   float* sq1 = sum1 + 64;
  float* sum2 = stats + 2 * 128;   float* sq2 = sum2 + 64;
  float* sumS = stats + 3 * 128;   float* sqS = sumS + 64;
  float* sc0 = sb + 0 * 128;  float* bb0 = sc0 + 64;
  float* sc1 = sb + 1 * 128;  float* bb1 = sc1 + 64;
  float* sc2 = sb + 2 * 128;  float* bb2 = sc2 + 64;
  float* scS = sb + 3 * 128;  float* bbS = scS + 64;

  const float invCntL  = 1.0f / (float)(Bv * Nv * Kv);   // 1/524288
  const float invCntSc = 1.0f / (float)(Bv * Nv);        // 1/32768

  efc_zero_stats<<<1, 512, 0, stream>>>(stats);
  efc_knn<<<(Bv * Nv) / 256, 256, 0, stream>>>(points, knn);
  efc_edge_init<<<(Bv * 4 * Mv) / 256, 256, 0, stream>>>(idxT, efT);
  efc_edge_scatter<<<(Bv * Pv) / 256, 256, 0, stream>>>(edges, edgef, idxT, efT);
  efc_idx_final<<<(Bv * Nv * Kv) / 256, 256, 0, stream>>>(idxT, knn, topI);

  efc_conv0<<<Bv * 128, 256, 0, stream>>>(features, W0, topI, efT, yA, sum0, sq0);
  efc_bn_finalize<<<1, 64, 0, stream>>>(sum0, sq0, g0, b0, invCntL, sc0, bb0);
  efc_conv12<<<Bv * 128, 256, 0, stream>>>(yA, W1, sc0, bb0, yB, sum1, sq1);
  efc_bn_finalize<<<1, 64, 0, stream>>>(sum1, sq1, g1, b1, invCntL, sc1, bb1);
  efc_conv12<<<Bv * 128, 256, 0, stream>>>(yB, W2, sc1, bb1, yA, sum2, sq2);
  efc_bn_finalize<<<1, 64, 0, stream>>>(sum2, sq2, g2, b2, invCntL, sc2, bb2);

  efc_sc<<<(Bv * Nv) / 128, 256, 0, stream>>>(features, scW, scBuf, sumS, sqS);
  efc_bn_finalize<<<1, 64, 0, stream>>>(sumS, sqS, scG, scB, invCntSc, scS, bbS);

  efc_final<<<(Bv * 64 * Nv) / 256, 256, 0, stream>>>(yA, sc2, bb2, scBuf, scS, bbS, out);
}